// Conv2d_20590073217670
// MI455X (gfx1250) — compile-verified
//
#include <hip/hip_runtime.h>

typedef float v2f __attribute__((ext_vector_type(2)));
typedef float v4f __attribute__((ext_vector_type(4)));
typedef float v8f __attribute__((ext_vector_type(8)));

#define C_IN   64
#define C_OUT  128
#define EDGE   64
#define KAREA  9
#define WSTRIDE (KAREA * C_OUT)        // 1152 floats between ci rows of weight
#define LDS_X  66                      // full 64-px row + 2 halo
#define LDS_CH_STRIDE (LDS_X * 3)      // 198 floats per input channel
#define LDS_TOTAL (C_IN * LDS_CH_STRIDE) // 12672 floats = 50.7 KB

// Implicit-GEMM conv, register-blocked: each block owns one (n, h) row:
// 64 output pixels x all 128 output channels. 8 wave32s arranged as
// 2 pixel-halves x 4 channel-quarters; each wave computes 32px x 32ch with
// 4 accumulators (2 A frags x 2 B frags), so every fragment feeds 2 WMMAs.
__global__ __launch_bounds__(256) void conv3x3_wmma_f32(
    const float* __restrict__ in,    // (32, 64, 64, 64) NCHW
    const float* __restrict__ wgt,   // (576, 128)  r = ci*9 + kh*3 + kw
    const float* __restrict__ bias,  // (128)
    float* __restrict__ out) {       // (32, 128, 64, 64) NCHW
  __shared__ float sIn[LDS_TOTAL];

  const int tid = threadIdx.x;
  const int t   = blockIdx.x;
  const int h0  = t & 63;
  const int n   = t >> 6;

  // ---- Stage zero-padded 3 x 66 x 64ch input row-tile into LDS
  const float* inN = in + n * (C_IN * EDGE * EDGE);
  for (int i = tid; i < LDS_TOTAL; i += 256) {
    int ci  = i / LDS_CH_STRIDE;
    int rem = i - ci * LDS_CH_STRIDE;
    int row = rem / LDS_X;
    int x   = rem - row * LDS_X;
    int gh  = h0 + row - 1;
    int gw  = x - 1;
    float v = 0.0f;
    if ((unsigned)gh < (unsigned)EDGE && (unsigned)gw < (unsigned)EDGE)
      v = inN[(ci * EDGE + gh) * EDGE + gw];
    sIn[i] = v;
  }
  __syncthreads();

  const int wv    = tid >> 5;
  const int lane  = tid & 31;
  const int hi    = lane >> 4;       // WMMA half: lanes 0-15 vs 16-31
  const int l16   = lane & 15;
  const int ph    = wv & 1;          // pixel half: w base = 32*ph
  const int cq    = wv >> 1;         // channel quarter: co base = 32*cq
  const int px0   = ph << 5;
  const int cb    = cq << 5;
  const int kbase = hi << 1;         // VGPR0 holds K=0 (lo half) / K=2 (hi half)

  v8f acc00 = {}, acc01 = {}, acc10 = {}, acc11 = {};

  for (int kidx = 0; kidx < KAREA; ++kidx) {
    const int kh = kidx / 3;
    const int kw = kidx - kh * 3;
    // A: pixel m -> LDS x = m + kw  (m = px0 + l16 [+16])
    const float* sA = sIn + kh * LDS_X + (px0 + l16 + kw);
    // B: weight row r = ci*9 + kidx, channel cb + l16 [+16]
    const float* wB = wgt + kidx * C_OUT + cb + l16;

    // Warm L2/L0 for the next (kh,kw) weight slab: gfx1250 global_prefetch_b8
    if (kidx + 1 < KAREA)
      __builtin_prefetch(wgt + (kidx + 1) * C_OUT + cb + l16, 0, 1);

#pragma unroll 4
    for (int ci0 = 0; ci0 < C_IN; ci0 += 4) {
      const int ck = ci0 + kbase;
      v2f a0, a1, b0, b1;
      a0.x = sA[ck * LDS_CH_STRIDE];
      a0.y = sA[(ck + 1) * LDS_CH_STRIDE];
      a1.x = sA[ck * LDS_CH_STRIDE + 16];
      a1.y = sA[(ck + 1) * LDS_CH_STRIDE + 16];
      b0.x = wB[ck * WSTRIDE];
      b0.y = wB[(ck + 1) * WSTRIDE];
      b1.x = wB[ck * WSTRIDE + 16];
      b1.y = wB[(ck + 1) * WSTRIDE + 16];
      acc00 = __builtin_amdgcn_wmma_f32_16x16x4_f32(false, a0, false, b0, (short)0, acc00, false, false);
      acc01 = __builtin_amdgcn_wmma_f32_16x16x4_f32(false, a0, false, b1, (short)0, acc01, false, false);
      acc10 = __builtin_amdgcn_wmma_f32_16x16x4_f32(false, a1, false, b0, (short)0, acc10, false, false);
      acc11 = __builtin_amdgcn_wmma_f32_16x16x4_f32(false, a1, false, b1, (short)0, acc11, false, false);
    }
  }

  // ---- Epilogue: D layout -> lane%16 = channel, VGPR r = pixel r + 8*hi.
  // Each lane's 8 acc elements are 8 contiguous w pixels -> 2x b128 stores.
  const float bv0 = bias[cb + l16];
  const float bv1 = bias[cb + 16 + l16];
  const v8f* accs[4] = {&acc00, &acc01, &acc10, &acc11};
#pragma unroll
  for (int pi = 0; pi < 2; ++pi) {
#pragma unroll
    for (int cj = 0; cj < 2; ++cj) {
      const v8f& a = *accs[pi * 2 + cj];
      const float bv = cj ? bv1 : bv0;
      const int co = cb + (cj << 4) + l16;
      const int w  = px0 + (pi << 4) + (hi << 3);
      float* outP = out + ((n * C_OUT + co) * EDGE + h0) * EDGE + w;
      v4f lo = {a[0] + bv, a[1] + bv, a[2] + bv, a[3] + bv};
      v4f hi4 = {a[4] + bv, a[5] + bv, a[6] + bv, a[7] + bv};
      *(v4f*)outP = lo;
      *((v4f*)outP + 1) = hi4;
    }
  }
}

extern "C" void kernel_launch(void* const* d_in, const int* in_sizes, int n_in,
                              void* d_out, int out_size, void* d_ws, size_t ws_size,
                              hipStream_t stream) {
  (void)in_sizes; (void)n_in; (void)out_size; (void)d_ws; (void)ws_size;
  const float* in   = (const float*)d_in[0];
  const float* wgt  = (const float*)d_in[1];
  const float* bias = (const float*)d_in[2];
  float* out = (float*)d_out;

  // 32 batches * 64 rows = 2048 blocks; each block = one full-width row,
  // 256 threads = 8 wave32s.
  dim3 grid(32 * 64);
  dim3 block(256);
  conv3x3_wmma_f32<<<grid, block, 0, stream>>>(in, wgt, bias, out);
}